// KAConvComparision_78056735637897
// MI455X (gfx1250) — compile-verified
//
#include <hip/hip_runtime.h>

// CDNA5 / gfx1250: wave32, WMMA 16x16 tiles.
typedef __attribute__((ext_vector_type(2))) float v2f;
typedef __attribute__((ext_vector_type(4))) float v4f;
typedef __attribute__((ext_vector_type(8))) float v8f;

__device__ __forceinline__ v2f make_v2f(float a, float b) {
    v2f r; r.x = a; r.y = b; return r;
}

// Problem constants (match reference setup_inputs()).
#define BBATCH 4
#define CC     16
#define HH     64
#define WW     64
#define FF     32
#define KK     3

// One wave handles a tile of 16 consecutive w positions (M = 16) x one
// f-tile of 16 filters (N = 16). 2048 waves total (1024 position tiles x
// 2 f-tiles) -> double the parallelism of the previous version and half
// the per-wave register pressure (one v8f accumulator).
//
// Per (channel c, kernel row a): one 18-wide row window of x is loaded
// (lanes 0..17, zero-padded at borders); the three bb shifts are produced
// with ds_bpermute_b32 instead of three redundant global loads.
//
// Inner polynomial contractions use V_WMMA_F32_16X16X4_F32:
//   P = WMMA(A[n=0..3], Bnum[n=0..3]); P = WMMA(A[n=4..5,pad], Bnum[n=4..5,pad], C=P)
//   Q = WMMA(A[m=1..4], Bden[m=1..4])
// Epilogue per tap: acc += P * rcp(1 + |Q|)  (abs folds into v_add source mod,
// rcp is a TRANS op that co-executes with the WMMA pipe).
__global__ __launch_bounds__(256) void pade_conv_wmma(
    const float* __restrict__ x,       // (B,C,H,W)
    const float* __restrict__ nums,    // (F,C,K,K,6)
    const float* __restrict__ denoms,  // (F,C,K,K,4)
    float* __restrict__ out)           // (B,F,H,W)
{
    const int lane  = threadIdx.x & 31;
    const int gw    = blockIdx.x * 8 + (threadIdx.x >> 5);  // 0..2047
    const int ftile = gw & 1;                                // which 16 filters
    const int tile  = gw >> 1;                               // 0..1023 position tiles

    const int w0 = (tile & 3) * 16;                          // W/16 = 4 strips
    const int h  = (tile >> 2) & (HH - 1);
    const int b  = tile >> 8;

    const int  laneM = lane & 15;    // M (position) / N (filter) index
    const bool hi    = lane >= 16;   // holds K rows 2,3 of the A fragment

    const int f = ftile * 16 + laneM;                // this lane's filter column
    const float* __restrict__ npF = nums   + (size_t)f * (CC * KK * KK) * 6;
    const float* __restrict__ dpF = denoms + (size_t)f * (CC * KK * KK) * 4;

    v8f acc = {};
    const v8f zc = {};

    // Row-window load slot: lanes 0..17 cover x[w0-1 .. w0+16] (zero-padded).
    const int  loadIdx = (lane < 18) ? lane : 17;    // lanes 18..31 redundant
    const int  wl      = w0 - 1 + loadIdx;
    const bool wlOK    = (unsigned)wl < (unsigned)WW;

    for (int c = 0; c < CC; ++c) {
        const float* __restrict__ xplane = x + (size_t)(b * CC + c) * (HH * WW);
        for (int a = 0; a < KK; ++a) {
            const int  hh  = h + a - 1;
            const bool hOK = (unsigned)hh < (unsigned)HH;
            const float* __restrict__ xrow = xplane + (size_t)(hOK ? hh : 0) * WW;

            float xload = 0.0f;
            if (hOK && wlOK) xload = xrow[wl];
            const int xbits = __float_as_int(xload);

#pragma unroll
            for (int bb = 0; bb < KK; ++bb) {
                // Shifted patch value via lane permute: src lane = laneM + bb (0..17).
                const float xv = __int_as_float(
                    __builtin_amdgcn_ds_bpermute((laneM + bb) << 2, xbits));

                const float x2 = xv * xv;
                const float x3 = x2 * xv;
                const float x4 = x2 * x2;
                const float x5 = x4 * xv;

                // A fragments (16x4 fp32 layout: lanes 0-15 hold K=0,1;
                // lanes 16-31 hold K=2,3; M = lane & 15).
                v2f aP0, aP1, aQ;
                if (!hi) {
                    aP0 = make_v2f(1.0f, xv);    // x^0, x^1
                    aP1 = make_v2f(x4,   x5);    // x^4, x^5
                    aQ  = make_v2f(xv,   x2);    // x^1, x^2
                } else {
                    aP0 = make_v2f(x2,   x3);    // x^2, x^3
                    aP1 = make_v2f(0.0f, 0.0f);  // pad rows 6,7
                    aQ  = make_v2f(x3,   x4);    // x^3, x^4
                }

                const int tap = (c * KK + a) * KK + bb;
                const float* np = npF + (size_t)tap * 6;
                const float* dp = dpF + (size_t)tap * 4;

                v2f bP0 = *(const v2f*)(np + (hi ? 2 : 0));
                v2f bP1 = hi ? make_v2f(0.0f, 0.0f) : *(const v2f*)(np + 4);
                v2f bQ  = *(const v2f*)(dp + (hi ? 2 : 0));

                v8f P = __builtin_amdgcn_wmma_f32_16x16x4_f32(
                    false, aP0, false, bP0, (short)0, zc, false, false);
                P = __builtin_amdgcn_wmma_f32_16x16x4_f32(
                    false, aP1, false, bP1, (short)0, P, false, false);
                v8f Q = __builtin_amdgcn_wmma_f32_16x16x4_f32(
                    false, aQ, false, bQ, (short)0, zc, false, false);

#pragma unroll
                for (int e = 0; e < 8; ++e)
                    acc[e] += P[e] * __builtin_amdgcn_rcpf(1.0f + __builtin_fabsf(Q[e]));
            }
        }
    }

    // C/D layout: lane l, element e -> (M = e + (l>=16 ? 8 : 0), N = l&15).
    // M is the w offset within the tile, N is the filter column.
    const int wbase = w0 + (hi ? 8 : 0);
    float* o = out + ((size_t)(b * FF + f) * HH + h) * WW + wbase;

    v4f s;
    s.x = acc[0]; s.y = acc[1]; s.z = acc[2]; s.w = acc[3];
    *(v4f*)(o) = s;
    s.x = acc[4]; s.y = acc[5]; s.z = acc[6]; s.w = acc[7];
    *(v4f*)(o + 4) = s;
}

extern "C" void kernel_launch(void* const* d_in, const int* in_sizes, int n_in,
                              void* d_out, int out_size, void* d_ws, size_t ws_size,
                              hipStream_t stream) {
    (void)in_sizes; (void)n_in; (void)out_size; (void)d_ws; (void)ws_size;
    const float* x      = (const float*)d_in[0];
    const float* nums   = (const float*)d_in[1];
    const float* denoms = (const float*)d_in[2];
    float* out = (float*)d_out;

    // 2048 wave-tiles (1024 position tiles x 2 f-tiles), 8 waves per block.
    pade_conv_wmma<<<256, 256, 0, stream>>>(x, nums, denoms, out);
}